// PatchCoherentSWDLoss_50362786512981
// MI455X (gfx1250) — compile-verified
//
#include <hip/hip_runtime.h>
#include <hip/hip_bf16.h>

typedef __attribute__((ext_vector_type(2))) float v2f;
typedef __attribute__((ext_vector_type(8))) float v8f;

#define NPATCH   8100      // 90*90
#define NROWPAD  8112      // 507 * 16
#define DDIM     147       // 3*7*7
#define DPAD     160       // padded row stride (zero-filled tail), 640B, 16B-aligned
#define KPAD     148       // 37 * 4
#define NPROJ    256
#define NSORT    8192      // sort width (pow2), entries >=8100 are +INF sentinels

#define XP_ELEMS (NROWPAD * DPAD)   // 1,297,920
#define R_ELEMS  (KPAD * NPROJ)     //    37,888
#define PT_ELEMS (NPROJ * NSORT)    // 2,097,152

// ---------------------------------------------------------------- init
__global__ __launch_bounds__(256) void k_init(float* pxT, float* pyT, double* acc) {
    long i = (long)blockIdx.x * 256 + threadIdx.x;
    const float INF = __int_as_float(0x7f800000);
    if (i < (long)PT_ELEMS) { pxT[i] = INF; pyT[i] = INF; }
    if (i == 0) *acc = 0.0;
}

// ------------------------------------------------------- patch extract
// x,y: [1,3,96,96]. Patch pr=(i*90+j); element t = c*49 + ph*7 + pw.
__global__ __launch_bounds__(160) void k_extract(const float* __restrict__ x,
                                                 const float* __restrict__ y,
                                                 float* __restrict__ xp,
                                                 float* __restrict__ yp) {
    int pr = blockIdx.x;          // 0..8111
    int t  = threadIdx.x;         // 0..159
    float vx = 0.f, vy = 0.f;
    if (pr < NPATCH && t < DDIM) {
        int i  = pr / 90, j = pr % 90;
        int c  = t / 49, rem = t % 49;
        int ph = rem / 7, pw = rem % 7;
        int src = c * 9216 + (i + ph) * 96 + (j + pw);
        vx = x[src]; vy = y[src];
    }
    xp[pr * DPAD + t] = vx;
    yp[pr * DPAD + t] = vy;
}

// ----------------------------------------------- normalize rand columns
// rand: [147, 256] row-major. r[k][p] = rand[k][p] / std(col p, ddof=1)
__global__ __launch_bounds__(256) void k_normrand(const float* __restrict__ rnd,
                                                  float* __restrict__ r) {
    int p = threadIdx.x;
    float mean = 0.f;
    for (int k = 0; k < DDIM; ++k) mean += rnd[k * NPROJ + p];
    mean *= (1.0f / DDIM);
    float var = 0.f;
    for (int k = 0; k < DDIM; ++k) {
        float d = rnd[k * NPROJ + p] - mean;
        var += d * d;
    }
    float inv = rsqrtf(var / (DDIM - 1));
    for (int k = 0; k < DDIM; ++k) r[k * NPROJ + p] = rnd[k * NPROJ + p] * inv;
    r[DDIM * NPROJ + p] = 0.f;    // zero pad row 147 (KPAD=148)
}

// ------------------------------------------------------- WMMA f32 GEMM
// D[m][n] = sum_k A[m][k] * R[k][n]; A: [NROWPAD, DPAD]; R: [KPAD, NPROJ]
// One wave per 16x16 tile. Output written transposed: Dst[n*NSORT + m].
__global__ __launch_bounds__(32) void k_gemm_wmma(const float* __restrict__ xp,
                                                  const float* __restrict__ yp,
                                                  const float* __restrict__ R,
                                                  float* __restrict__ pxT,
                                                  float* __restrict__ pyT) {
    const float* A   = blockIdx.z ? yp : xp;
    float*       Dst = blockIdx.z ? pyT : pxT;

    int l     = threadIdx.x;               // 0..31
    int m0    = blockIdx.x * 16;           // 507 tiles
    int n0    = blockIdx.y * 16;           // 16 tiles
    int half  = l >> 4;                    // 0 or 1
    int lsub  = l & 15;
    int kbase = half * 2;                  // A/B: VGPR0 -> K=kbase, VGPR1 -> K=kbase+1
    int mrow  = m0 + lsub;
    int ncol  = n0 + lsub;

    v8f acc = {};
    const float* arow = A + (long)mrow * DPAD;
    for (int k0 = 0; k0 < KPAD; k0 += 4) {
        v2f a, b;
        a.x = arow[k0 + kbase];
        a.y = arow[k0 + kbase + 1];
        b.x = R[(k0 + kbase) * NPROJ + ncol];
        b.y = R[(k0 + kbase + 1) * NPROJ + ncol];
        acc = __builtin_amdgcn_wmma_f32_16x16x4_f32(false, a, false, b,
                                                    (short)0, acc, false, false);
    }

    // C/D layout: VGPR v holds row m = m0 + half*8 + v, col = n0 + lsub
    int mbase = m0 + half * 8;
#pragma unroll
    for (int v = 0; v < 8; ++v) {
        int m = mbase + v;
        if (m < NPATCH) Dst[(long)ncol * NSORT + m] = acc[v];
    }
}

// ---------------------------------------------- bitonic argsort (LDS)
// Packed 64-bit: [order-preserving key bits : 32 | index : 32].
// Tie-break by index == stable argsort (matches jnp.argsort).
// +INF sentinels (entries >= 8100) sort last, so ranks < 8100 carry
// indices < 8100.
__global__ __launch_bounds__(1024) void k_argsort(const float* __restrict__ pxT,
                                                  const float* __restrict__ pyT,
                                                  int* __restrict__ ix,
                                                  int* __restrict__ iy) {
    __shared__ unsigned long long kv[NSORT];   // 64 KB
    int p = blockIdx.x;
    const float* src = blockIdx.y ? pyT : pxT;
    int*         dst = blockIdx.y ? iy  : ix;
    int tid = threadIdx.x;

    for (int i = tid; i < NSORT; i += 1024) {
        unsigned u = __float_as_uint(src[(long)p * NSORT + i]);
        u = (u & 0x80000000u) ? ~u : (u | 0x80000000u);   // monotonic map
        kv[i] = ((unsigned long long)u << 32) | (unsigned)i;
    }
    __syncthreads();

    for (unsigned k = 2; k <= NSORT; k <<= 1) {
        for (unsigned j = k >> 1; j > 0; j >>= 1) {
            for (unsigned t = tid; t < NSORT; t += 1024) {
                unsigned i = t, x = i ^ j;
                if (x > i) {
                    bool up = ((i & k) == 0);              // ascending block
                    unsigned long long a = kv[i], b = kv[x];
                    if (up ? (a > b) : (a < b)) { kv[i] = b; kv[x] = a; }
                }
            }
            __syncthreads();
        }
    }

    for (int i = tid; i < NSORT; i += 1024)
        dst[(long)p * NSORT + i] = (int)(kv[i] & 0xFFFFFFFFu);
}

// --------------------------------------------------- gather + L1 reduce
// grid: (64 rank-chunks, 256 projections); block 256 = 8 waves.
// Each wave handles 16 ranks as 4 iterations x 4 concurrent ranks;
// 8 lanes per rank, 5 float4 loads per row per lane (covers DPAD=160;
// zero-padded tails contribute |0-0| = 0).
__global__ __launch_bounds__(256) void k_gather_l1(const float* __restrict__ xp,
                                                   const float* __restrict__ yp,
                                                   const int* __restrict__ ix,
                                                   const int* __restrict__ iy,
                                                   double* __restrict__ acc) {
    int p    = blockIdx.y;
    int wave = threadIdx.x >> 5;
    int lane = threadIdx.x & 31;
    int g    = lane >> 3;                  // rank-within-quad: 0..3
    int sub  = lane & 7;                   // sublane within rank group
    int nbase = blockIdx.x * 128 + wave * 16;

    const int* ixp = ix + (long)p * NSORT;
    const int* iyp = iy + (long)p * NSORT;

    float s = 0.f;
#pragma unroll
    for (int t = 0; t < 4; ++t) {
        int n  = nbase + t * 4 + g;
        int nc = (n < NPATCH) ? n : (NPATCH - 1);
        float msk = (n < NPATCH) ? 1.f : 0.f;
        int a = ixp[nc];
        int b = iyp[nc];
        const float4* xr = (const float4*)(xp + (long)a * DPAD);
        const float4* yr = (const float4*)(yp + (long)b * DPAD);

        if (t < 3) {                       // prefetch next iteration's rows
            int n2 = nbase + t * 4 + 4 + g;
            int n2c = (n2 < NPATCH) ? n2 : (NPATCH - 1);
            __builtin_prefetch(xp + (long)ixp[n2c] * DPAD, 0, 0);
            __builtin_prefetch(yp + (long)iyp[n2c] * DPAD, 0, 0);
        }

        float r = 0.f;
#pragma unroll
        for (int c = 0; c < 5; ++c) {      // 8 sublanes * 5 float4 = 160 floats
            float4 xv = xr[sub + c * 8];
            float4 yv = yr[sub + c * 8];
            r += fabsf(xv.x - yv.x) + fabsf(xv.y - yv.y) +
                 fabsf(xv.z - yv.z) + fabsf(xv.w - yv.w);
        }
        s += msk * r;
    }

    // wave32 reduction
    for (int off = 16; off; off >>= 1) s += __shfl_xor(s, off, 32);

    __shared__ float wsum[8];
    if (lane == 0) wsum[wave] = s;
    __syncthreads();
    if (threadIdx.x == 0) {
        float bs = 0.f;
        for (int w = 0; w < 8; ++w) bs += wsum[w];
        atomicAdd(acc, (double)bs);        // GLOBAL_ATOMIC_ADD_F64
    }
}

// -------------------------------------------------------------- finalize
__global__ __launch_bounds__(32) void k_finalize(const double* __restrict__ acc,
                                                 float* __restrict__ out) {
    if (threadIdx.x == 0 && blockIdx.x == 0) {
        double denom = (double)NPROJ * (double)NPATCH * (double)DDIM;
        out[0] = (float)(*acc / denom);
    }
}

extern "C" void kernel_launch(void* const* d_in, const int* in_sizes, int n_in,
                              void* d_out, int out_size, void* d_ws, size_t ws_size,
                              hipStream_t stream) {
    const float* x    = (const float*)d_in[0];   // [1,3,96,96]
    const float* y    = (const float*)d_in[1];   // [1,3,96,96]
    const float* rnd  = (const float*)d_in[2];   // [147,256]
    float* out = (float*)d_out;

    double* acc  = (double*)d_ws;
    float*  base = (float*)d_ws + 16;            // 64B past accumulator
    float*  xp   = base;
    float*  yp   = xp + XP_ELEMS;
    float*  rn   = yp + XP_ELEMS;
    float*  pxT  = rn + R_ELEMS;
    float*  pyT  = pxT + PT_ELEMS;
    int*    ix   = (int*)(pyT + PT_ELEMS);
    int*    iy   = ix + PT_ELEMS;

    // 1) sentinel-init pxT/pyT (+INF) and zero accumulator
    k_init<<<(PT_ELEMS + 255) / 256, 256, 0, stream>>>(pxT, pyT, acc);
    // 2) patch extraction (zero-padded rows/cols)
    k_extract<<<NROWPAD, DPAD, 0, stream>>>(x, y, xp, yp);
    // 3) normalize projection matrix
    k_normrand<<<1, NPROJ, 0, stream>>>(rnd, rn);
    // 4) projections via f32 WMMA (one wave per 16x16 tile, z selects x/y)
    k_gemm_wmma<<<dim3(NROWPAD / 16, NPROJ / 16, 2), 32, 0, stream>>>(xp, yp, rn, pxT, pyT);
    // 5) 512 packed-64-bit bitonic argsorts in LDS (stable, like jnp.argsort)
    k_argsort<<<dim3(NPROJ, 2), 1024, 0, stream>>>(pxT, pyT, ix, iy);
    // 6) rank-matched gather + L1 accumulation (float4 rows, prefetch)
    k_gather_l1<<<dim3(64, NPROJ), 256, 0, stream>>>(xp, yp, ix, iy, acc);
    // 7) scalar mean
    k_finalize<<<1, 32, 0, stream>>>(acc, out);
}